// DeformableDWConv_17557826306545
// MI455X (gfx1250) — compile-verified
//
#include <hip/hip_runtime.h>
#include <math.h>
#include <stdint.h>

#define BB   8
#define CC   128
#define HH   96
#define WW   96
#define HWW  (HH * WW)      // 9216
#define TAPS 64

// ---------------------------------------------------------------------------
// K0a: transpose depthwise weight (1, C, 8, 8) -> wt_t[t * C + c]
// ---------------------------------------------------------------------------
__global__ void k0_wt_transpose(const float* __restrict__ weight,
                                float* __restrict__ wt_t) {
    int i = blockIdx.x * 256 + threadIdx.x;       // 0 .. 8191
    if (i < TAPS * CC) {
        int t = i >> 7;          // tap
        int c = i & 127;         // channel
        wt_t[i] = weight[c * TAPS + t];
    }
}

// ---------------------------------------------------------------------------
// K0b: fold BatchNorm into per-channel affine: scale = g/sqrt(v+eps),
//      shift = beta - mean*scale
// ---------------------------------------------------------------------------
__global__ void k0_bn_prep(const float* __restrict__ g,
                           const float* __restrict__ be,
                           const float* __restrict__ m,
                           const float* __restrict__ v,
                           float* __restrict__ scale,
                           float* __restrict__ shift) {
    int c = threadIdx.x;                          // 0..127
    float s = g[c] / sqrtf(v[c] + 1e-5f);
    scale[c] = s;
    shift[c] = be[c] - m[c] * s;
}

// ---------------------------------------------------------------------------
// K1: NCHW -> NHWC transpose of x via LDS tile (coalesced both ways)
// ---------------------------------------------------------------------------
__global__ void k1_x_transpose(const float* __restrict__ x,
                               float* __restrict__ x_t) {
    __shared__ float tile[32][33];
    int b  = blockIdx.z;
    int c0 = blockIdx.y * 32;
    int p0 = blockIdx.x * 32;
    int tx = threadIdx.x, ty = threadIdx.y;
    const float* xb  = x   + (size_t)b * CC * HWW;
    float*       xtb = x_t + (size_t)b * HWW * CC;
#pragma unroll
    for (int k = 0; k < 4; ++k)
        tile[ty + 8 * k][tx] = xb[(size_t)(c0 + ty + 8 * k) * HWW + (p0 + tx)];
    __syncthreads();
#pragma unroll
    for (int k = 0; k < 4; ++k)
        xtb[(size_t)(p0 + ty + 8 * k) * CC + (c0 + tx)] = tile[tx][ty + 8 * k];
}

// ---------------------------------------------------------------------------
// K2: offset prediction = depthwise 3x3 conv of x (groups = C), zero padded.
//     off channel c -> comp = c/64, tap = c%64.
//     Store interleaved per pixel: off_ws[pix*128 + 2*tap + comp]
// ---------------------------------------------------------------------------
__global__ void k2_offset_conv(const float* __restrict__ x_t,
                               const float* __restrict__ offw,   // (128,1,3,3)
                               const float* __restrict__ offb,   // (128,)
                               float* __restrict__ off_ws) {
    int tid   = threadIdx.x;
    int local = tid >> 7;                  // pixel within block (0..1)
    int c     = tid & 127;
    int pix   = blockIdx.x * 2 + local;    // flat (b, h, w)
    int b     = pix / HWW;
    int p     = pix - b * HWW;
    int h     = p / WW;
    int w     = p - h * WW;

    float acc = offb[c];
    const float* wr = offw + c * 9;
    const float* xb = x_t + (size_t)b * HWW * CC;
#pragma unroll
    for (int i = -1; i <= 1; ++i) {
        int hh = h + i;
        if (hh < 0 || hh >= HH) continue;
#pragma unroll
        for (int j = -1; j <= 1; ++j) {
            int ww2 = w + j;
            if (ww2 < 0 || ww2 >= WW) continue;
            acc += xb[(hh * WW + ww2) * CC + c] * wr[(i + 1) * 3 + (j + 1)];
        }
    }
    off_ws[(size_t)pix * 128 + ((c & 63) << 1) + (c >> 6)] = acc;
}

// ---------------------------------------------------------------------------
// K3: deformable gather + depthwise accumulate + fused BN affine + exact GELU
//     wave32 <-> one output pixel, lane <-> 4 channels (float4 path).
//     Branch-free OOB: clamped loads + validity-masked bilinear weights.
// ---------------------------------------------------------------------------
__launch_bounds__(256, 4)
__global__ void k3_deform(const float* __restrict__ x_t,
                          const float* __restrict__ off_ws,
                          const float* __restrict__ wt_t,
                          const float* __restrict__ bnscale,
                          const float* __restrict__ bnshift,
                          float* __restrict__ out) {
    __shared__ __align__(16) float wt_s[TAPS * CC];   // 32 KB  [tap][c]
    __shared__ __align__(16) float off_s[8][CC];      //  4 KB  per-pixel (dx,dy)x64

    int tid  = threadIdx.x;
    int wave = tid >> 5;
    int lane = tid & 31;
    int c4   = lane << 2;

    // --- CDNA5 async copy: stage 64x128 tap-weight matrix into LDS --------
    unsigned lds_base = (unsigned)(uintptr_t)&wt_s[0];
#pragma unroll
    for (int k = 0; k < 8; ++k) {
        unsigned e     = (unsigned)(k * 256 + tid);      // float4 element index
        unsigned laddr = lds_base + e * 16u;
        const float* g = wt_t + (size_t)e * 4;
        asm volatile("global_load_async_to_lds_b128 %0, %1, off"
                     :: "v"(laddr), "v"(g)
                     : "memory");
    }

    int pix = blockIdx.x * 8 + wave;       // flat (b, h, w); grid covers exactly
    int b   = pix / HWW;
    int p   = pix - b * HWW;
    int h   = p / WW;
    int w   = p - h * WW;
    const float* __restrict__ xb = x_t + (size_t)b * HWW * CC;  // 32-bit indexable

    // stage this pixel's 64 (dx,dy) pairs into LDS (coalesced float4)
    float4 o4 = *(const float4*)&off_ws[(size_t)pix * 128 + c4];
    *(float4*)&off_s[wave][c4] = o4;

    // warm the gather footprint (emits global_prefetch_b8)
    __builtin_prefetch(&xb[p * CC + c4], 0, 3);

    asm volatile("s_wait_asynccnt 0" ::: "memory");
    __syncthreads();

    const float sc = 96.0f / 95.0f;        // W/(W-1) == H/(H-1)
    const float fw = (float)w, fh = (float)h;
    float4 acc = make_float4(0.f, 0.f, 0.f, 0.f);

#pragma unroll 4
    for (int t = 0; t < TAPS; ++t) {
        float2 d2 = *(const float2*)&off_s[wave][2 * t];
        float kx = (float)(t & 7)  - 3.5f;
        float ky = (float)(t >> 3) - 3.5f;
        float ix = (fw + kx + d2.x) * sc - 0.5f;
        float iy = (fh + ky + d2.y) * sc - 0.5f;
        float x0f = floorf(ix), y0f = floorf(iy);
        float fx = ix - x0f,    fy = iy - y0f;
        int   x0 = (int)x0f,    y0 = (int)y0f;
        int   x1 = x0 + 1,      y1 = y0 + 1;

        // validity masks (branch-free); clamped coordinates for safe loads
        float vx0 = ((unsigned)x0 < (unsigned)WW) ? 1.f : 0.f;
        float vx1 = ((unsigned)x1 < (unsigned)WW) ? 1.f : 0.f;
        float vy0 = ((unsigned)y0 < (unsigned)HH) ? 1.f : 0.f;
        float vy1 = ((unsigned)y1 < (unsigned)HH) ? 1.f : 0.f;
        int xc0 = min(max(x0, 0), WW - 1), xc1 = min(max(x1, 0), WW - 1);
        int yc0 = min(max(y0, 0), HH - 1), yc1 = min(max(y1, 0), HH - 1);

        float w00 = (1.f - fx) * (1.f - fy) * (vx0 * vy0);
        float w01 = fx * (1.f - fy)         * (vx1 * vy0);
        float w10 = (1.f - fx) * fy         * (vx0 * vy1);
        float w11 = fx * fy                 * (vx1 * vy1);

        int r0 = yc0 * WW, r1 = yc1 * WW;
        float4 v00 = *(const float4*)&xb[(r0 + xc0) * CC + c4];
        float4 v01 = *(const float4*)&xb[(r0 + xc1) * CC + c4];
        float4 v10 = *(const float4*)&xb[(r1 + xc0) * CC + c4];
        float4 v11 = *(const float4*)&xb[(r1 + xc1) * CC + c4];

        float4 wt4 = *(const float4*)&wt_s[t * CC + c4];
        acc.x = fmaf(wt4.x, w00 * v00.x + w01 * v01.x + w10 * v10.x + w11 * v11.x, acc.x);
        acc.y = fmaf(wt4.y, w00 * v00.y + w01 * v01.y + w10 * v10.y + w11 * v11.y, acc.y);
        acc.z = fmaf(wt4.z, w00 * v00.z + w01 * v01.z + w10 * v10.z + w11 * v11.z, acc.z);
        acc.w = fmaf(wt4.w, w00 * v00.w + w01 * v01.w + w10 * v10.w + w11 * v11.w, acc.w);
    }

    // --- folded BatchNorm affine + exact GELU -----------------------------
    float4 s4 = *(const float4*)&bnscale[c4];
    float4 h4 = *(const float4*)&bnshift[c4];

    const float is2 = 0.70710678118654752440f;
    float r[4];
    {
        float t0 = fmaf(acc.x, s4.x, h4.x);
        r[0] = 0.5f * t0 * (1.0f + erff(t0 * is2));
    }
    {
        float t0 = fmaf(acc.y, s4.y, h4.y);
        r[1] = 0.5f * t0 * (1.0f + erff(t0 * is2));
    }
    {
        float t0 = fmaf(acc.z, s4.z, h4.z);
        r[2] = 0.5f * t0 * (1.0f + erff(t0 * is2));
    }
    {
        float t0 = fmaf(acc.w, s4.w, h4.w);
        r[3] = 0.5f * t0 * (1.0f + erff(t0 * is2));
    }

#pragma unroll
    for (int j = 0; j < 4; ++j)
        out[(size_t)(b * CC + c4 + j) * HWW + p] = r[j];
}

// ---------------------------------------------------------------------------
// Launch
// ---------------------------------------------------------------------------
extern "C" void kernel_launch(void* const* d_in, const int* in_sizes, int n_in,
                              void* d_out, int out_size, void* d_ws, size_t ws_size,
                              hipStream_t stream) {
    const float* x       = (const float*)d_in[0];
    const float* offw    = (const float*)d_in[1];
    const float* offb    = (const float*)d_in[2];
    const float* weight  = (const float*)d_in[3];
    const float* gamma   = (const float*)d_in[4];
    const float* beta    = (const float*)d_in[5];
    const float* mean    = (const float*)d_in[6];
    const float* var     = (const float*)d_in[7];
    float* out = (float*)d_out;

    char* ws = (char*)d_ws;
    const size_t xbytes = (size_t)BB * CC * HWW * sizeof(float);   // 37,748,736
    float* x_t     = (float*)(ws);
    float* off_ws  = (float*)(ws + xbytes);
    float* wt_t    = (float*)(ws + 2 * xbytes);                    // 32 KB
    float* bnscale = (float*)(ws + 2 * xbytes + 32768);            // 512 B
    float* bnshift = (float*)(ws + 2 * xbytes + 32768 + 512);      // 512 B

    // K0a: weight transpose (8192 elems)
    k0_wt_transpose<<<32, 256, 0, stream>>>(weight, wt_t);

    // K0b: fold BN into per-channel affine
    k0_bn_prep<<<1, 128, 0, stream>>>(gamma, beta, mean, var, bnscale, bnshift);

    // K1: x NCHW -> NHWC
    dim3 g1(HWW / 32, CC / 32, BB);   // (288, 4, 8)
    dim3 b1(32, 8);
    k1_x_transpose<<<g1, b1, 0, stream>>>(x, x_t);

    // K2: offset depthwise 3x3 conv (2 pixels x 128 ch per block)
    k2_offset_conv<<<(BB * HWW) / 2, 256, 0, stream>>>(x_t, offw, offb, off_ws);

    // K3: main deformable gather + BN + GELU (8 pixels x 4-ch lanes per block)
    k3_deform<<<(BB * HWW) / 8, 256, 0, stream>>>(x_t, off_ws, wt_t,
                                                  bnscale, bnshift, out);
}